// RoiPooling_70531952935075
// MI455X (gfx1250) — compile-verified
//
#include <hip/hip_runtime.h>
#include <cstdint>

typedef unsigned int u32x4 __attribute__((ext_vector_type(4)));
typedef unsigned int u32x8 __attribute__((ext_vector_type(8)));

namespace {
constexpr int IMG_B = 2, IMG_H = 38, IMG_W = 38, IMG_C = 512, NROI = 64;
constexpr int KMAX       = 19;               // max ROI width/height (W/2)
constexpr int CSPLIT     = 4;                // channel slices per ROI
constexpr int CH_PER_BLK = IMG_C / CSPLIT;   // 128 channels per 1-wave block
constexpr int BUF_FLOATS = KMAX * CH_PER_BLK;    // 2432 floats per row buffer
constexpr int BUF_BYTES  = BUF_FLOATS * 4;       // 9728 B (multiple of 512)
}

// One wave32 per (ROI, 128-channel slice). Each ROI row is staged into LDS by
// a single Tensor Data Mover descriptor (2-D tile: 128 channels x tw pixels,
// dim0 stride = 512 elements), double-buffered via the in-order TENSORcnt.
// Reduction: ds_load_b128 + v_max_num_f32.
__global__ __launch_bounds__(32) void roi_maxpool_tdm(
    const float* __restrict__ img, const int* __restrict__ rois,
    float* __restrict__ out)
{
    __shared__ __align__(1024) float smem[2 * BUF_FLOATS];   // 19456 B

    const int t     = threadIdx.x;           // lane 0..31 (wave32)
    const int br    = blockIdx.x;            // flattened (b, roi)
    const int b     = br >> 6;               // / NROI
    const int choff = blockIdx.y * CH_PER_BLK;

    const int4 box = reinterpret_cast<const int4*>(rois)[br];  // uniform s_load
    const int ow = box.x, oh = box.y, tw = box.z, th = box.w;

    // Tile start for (b, oh, ow) at this block's channel slice (wave-uniform).
    const uint64_t gbase = (uint64_t)(uintptr_t)img
        + (uint64_t)((((b * IMG_H + oh) * IMG_W) + ow) * IMG_C + choff) * 4ull;
    const uint64_t row_pitch = (uint64_t)IMG_W * IMG_C * 4ull;  // 77824 B
    // Low 32 bits of a flat shared pointer == LDS byte offset on gfx1250.
    const uint32_t lds_base = (uint32_t)(uintptr_t)(&smem[0]);

    // Issue one TDM 2-D tile load: [tile_dim0=128 ch] x [tile_dim1=tw pixels],
    // dim0 stride 512 elements (2 KB), packed contiguously into LDS buffer p.
    auto stage_row = [&](int iy, int p) {
        const uint64_t ga = gbase + (uint64_t)iy * row_pitch;
        u32x4 g0;
        g0[0] = 1u;                                   // count=1: valid user D#
        g0[1] = lds_base + (uint32_t)p * BUF_BYTES;   // lds_addr (bytes)
        g0[2] = (uint32_t)ga;                         // global_addr[31:0]
        g0[3] = (uint32_t)(ga >> 32) | 0x80000000u;   // addr[56:32] | type=2
        u32x8 g1;
        g1[0] = 0x00020000u;                  // workgroup_mask=0, data_size=4B
        g1[1] = 512u << 16;                   // tensor_dim0 = 512  (no OOB)
        g1[2] = 64u  << 16;                   // tensor_dim1 = 64   (no OOB)
        g1[3] = (uint32_t)CH_PER_BLK << 16;   // tile_dim0 = 128 elements
        g1[4] = (uint32_t)tw;                 // tile_dim1 = tw, tile_dim2 = 0
        g1[5] = 512u;                         // tensor_dim0_stride = 512 elems
        g1[6] = 0u;
        g1[7] = 0u;
        asm volatile("tensor_load_to_lds %0, %1" :: "s"(g0), "s"(g1) : "memory");
    };

    const float ninf = -__builtin_inff();
    float4 acc = make_float4(ninf, ninf, ninf, ninf);

    stage_row(0, 0);
    for (int iy = 0; iy < th; ++iy) {
        const int cur = iy & 1;
        if (iy + 1 < th) {
            // WAR: drain ds reads of the buffer about to be refilled, then
            // start the next row's DMA while this row is reduced.
            asm volatile("s_wait_dscnt 0x0" ::: "memory");
            stage_row(iy + 1, cur ^ 1);
            // TDM ops complete in order: <=1 outstanding => row iy landed.
            asm volatile("s_wait_tensorcnt 0x1" ::: "memory");
        } else {
            asm volatile("s_wait_tensorcnt 0x0" ::: "memory");
        }
        const float* lp = &smem[cur * BUF_FLOATS + t * 4];
        for (int x = 0; x < tw; ++x) {
            const float4 v = *reinterpret_cast<const float4*>(lp + x * CH_PER_BLK);
            acc.x = fmaxf(acc.x, v.x);
            acc.y = fmaxf(acc.y, v.y);
            acc.z = fmaxf(acc.z, v.z);
            acc.w = fmaxf(acc.w, v.w);
        }
    }

    *reinterpret_cast<float4*>(out + (size_t)br * IMG_C + choff + t * 4) = acc;
}

extern "C" void kernel_launch(void* const* d_in, const int* /*in_sizes*/, int /*n_in*/,
                              void* d_out, int /*out_size*/, void* /*d_ws*/, size_t /*ws_size*/,
                              hipStream_t stream) {
    const float* img  = (const float*)d_in[0];
    const int*   rois = (const int*)d_in[1];
    float*       out  = (float*)d_out;

    dim3 grid(IMG_B * NROI, CSPLIT, 1);   // 128 ROIs x 4 channel slices
    roi_maxpool_tdm<<<grid, dim3(32, 1, 1), 0, stream>>>(img, rois, out);
}